// MiniBAInternal_20315195310085
// MI455X (gfx1250) — compile-verified
//
#include <hip/hip_runtime.h>
#include <math.h>

#define NPX   16384
#define NCX   8
#define NPAIRX (NPX*NCX)      // 131072
#define R2X   (NPAIRX*2)      // 262144
#define NCPX  73
#define PADX  80
#define NBX   1024
#define EPX   1e-6f
#define LM0X  0.001f
#define KFX   2.0f

typedef float v2f __attribute__((ext_vector_type(2)));
typedef float v8f __attribute__((ext_vector_type(8)));

// ---------------- workspace layout (floats) ----------------
constexpr size_t O_CUR   = 0;                         // rs6d(48)+ts(24)+f(1)+xyz(49152) = 49225
constexpr size_t O_TMP   = O_CUR   + 49225;
constexpr size_t O_CAMD  = O_TMP   + 49225;           // NC*64
constexpr size_t O_CAMDT = O_CAMD  + 512;
constexpr size_t O_RIN   = O_CAMDT + 512;             // R2
constexpr size_t O_RW    = O_RIN   + R2X;
constexpr size_t O_W     = O_RW    + R2X;
constexpr size_t O_ERR   = O_W     + R2X;             // NPAIR
constexpr size_t O_MASK  = O_ERR   + NPAIRX;
constexpr size_t O_JC    = O_MASK  + NPAIRX;          // NPAIR*20
constexpr size_t O_JX    = O_JC    + (size_t)NPAIRX*20; // NPAIR*6
constexpr size_t O_JTJPC = O_JX    + (size_t)NPAIRX*6;  // NC*256
constexpr size_t O_JTJ   = O_JTJPC + 2048;            // 80*80
constexpr size_t O_S     = O_JTJ   + 6400;
constexpr size_t O_GC    = O_S     + 6400;            // 80
constexpr size_t O_BDGX  = O_GC    + 80;
constexpr size_t O_RHS   = O_BDGX  + 80;
constexpr size_t O_DCAM  = O_RHS   + 80;
constexpr size_t O_INV3  = O_DCAM  + 80;              // NP*9
constexpr size_t O_GX    = O_INV3  + (size_t)NPX*9;   // NP*3
constexpr size_t O_E3    = O_GX    + (size_t)NPX*3;   // NP*3*80
constexpr size_t O_DXYZ  = O_E3    + (size_t)NPX*3*PADX;
constexpr size_t O_SCAL  = O_DXYZ  + (size_t)NPX*3;   // 16: 0=lm 1=maxbits 2=med 3=mad 4=c 5=mseo 6=msen 7=sm 8=valid 9=maxd
constexpr size_t O_HIST  = O_SCAL  + 16;              // 1024 uint

// output layout (floats)
constexpr size_t OO_XFER = 49225;    // params block (rs+ts+f+xyz)
constexpr size_t OO_RFIN = 49225;
constexpr size_t OO_RINIT= OO_RFIN + R2X;
constexpr size_t OO_MASK = OO_RINIT + R2X;

// ---------------- small init ----------------
__global__ void k_init(float* scal) {
  if (threadIdx.x == 0) scal[0] = LM0X;
}

// ---------------- camera prep: R columns + Gram-Schmidt derivative tensors ----
__device__ inline void gs_b1b2(const float* r6, float* b1, float* b2) {
  float a1[3] = {r6[0], r6[2], r6[4]};
  float a2[3] = {r6[1], r6[3], r6[5]};
  float in1 = rsqrtf(a1[0]*a1[0] + a1[1]*a1[1] + a1[2]*a1[2]);
  b1[0]=a1[0]*in1; b1[1]=a1[1]*in1; b1[2]=a1[2]*in1;
  float d = b1[0]*a2[0]+b1[1]*a2[1]+b1[2]*a2[2];
  float u0=a2[0]-d*b1[0], u1=a2[1]-d*b1[1], u2=a2[2]-d*b1[2];
  float inu = rsqrtf(u0*u0+u1*u1+u2*u2);
  b2[0]=u0*inu; b2[1]=u1*inu; b2[2]=u2*inu;
}

__global__ void k_camprep(const float* params, float* camD) {
  int c = threadIdx.x;
  if (c >= NCX) return;
  const float* r = params + c*6;
  const float* t = params + 48 + c*3;
  float a1[3] = {r[0], r[2], r[4]};
  float a2[3] = {r[1], r[3], r[5]};
  float n1 = sqrtf(a1[0]*a1[0]+a1[1]*a1[1]+a1[2]*a1[2]);
  float in1 = 1.f/n1;
  float b1[3] = {a1[0]*in1, a1[1]*in1, a1[2]*in1};
  float d = b1[0]*a2[0]+b1[1]*a2[1]+b1[2]*a2[2];
  float u[3] = {a2[0]-d*b1[0], a2[1]-d*b1[1], a2[2]-d*b1[2]};
  float nu = sqrtf(u[0]*u[0]+u[1]*u[1]+u[2]*u[2]);
  float inu = 1.f/nu;
  float b2[3] = {u[0]*inu, u[1]*inu, u[2]*inu};
  float b3[3] = {b1[1]*b2[2]-b1[2]*b2[1], b1[2]*b2[0]-b1[0]*b2[2], b1[0]*b2[1]-b1[1]*b2[0]};
  float D1[9], dU1[9], dU2[9], P2[9], D2a1[9], D2a2[9], D3a1[9], D3a2[9];
  for (int i=0;i<3;i++) for (int j=0;j<3;j++)
    D1[i*3+j] = (((i==j)?1.f:0.f) - b1[i]*b1[j]) * in1;
  for (int j=0;j<3;j++) {
    float s = D1[0*3+j]*a2[0] + D1[1*3+j]*a2[1] + D1[2*3+j]*a2[2];
    for (int i=0;i<3;i++) dU1[i*3+j] = -(s*b1[i] + d*D1[i*3+j]);
  }
  for (int i=0;i<3;i++) for (int j=0;j<3;j++)
    dU2[i*3+j] = ((i==j)?1.f:0.f) - b1[i]*b1[j];
  for (int i=0;i<3;i++) for (int j=0;j<3;j++)
    P2[i*3+j] = (((i==j)?1.f:0.f) - b2[i]*b2[j]) * inu;
  for (int i=0;i<3;i++) for (int j=0;j<3;j++) {
    float s1=0.f, s2=0.f;
    for (int k=0;k<3;k++) { s1 += P2[i*3+k]*dU1[k*3+j]; s2 += P2[i*3+k]*dU2[k*3+j]; }
    D2a1[i*3+j]=s1; D2a2[i*3+j]=s2;
  }
  for (int j=0;j<3;j++) {
    float c1[3] = {D1[0*3+j], D1[1*3+j], D1[2*3+j]};
    float c2[3] = {D2a1[0*3+j], D2a1[1*3+j], D2a1[2*3+j]};
    float c3[3] = {D2a2[0*3+j], D2a2[1*3+j], D2a2[2*3+j]};
    D3a1[0*3+j] = (c1[1]*b2[2]-c1[2]*b2[1]) + (b1[1]*c2[2]-b1[2]*c2[1]);
    D3a1[1*3+j] = (c1[2]*b2[0]-c1[0]*b2[2]) + (b1[2]*c2[0]-b1[0]*c2[2]);
    D3a1[2*3+j] = (c1[0]*b2[1]-c1[1]*b2[0]) + (b1[0]*c2[1]-b1[1]*c2[0]);
    D3a2[0*3+j] = b1[1]*c3[2]-b1[2]*c3[1];
    D3a2[1*3+j] = b1[2]*c3[0]-b1[0]*c3[2];
    D3a2[2*3+j] = b1[0]*c3[1]-b1[1]*c3[0];
  }
  float* o = camD + c*64;
  for (int i=0;i<3;i++) { o[i]=b1[i]; o[3+i]=b2[i]; o[6+i]=b3[i]; o[9+i]=t[i]; }
  for (int i=0;i<9;i++) { o[12+i]=D1[i]; o[21+i]=D2a1[i]; o[30+i]=D2a2[i]; o[39+i]=D3a1[i]; o[48+i]=D3a2[i]; }
}

// ---------------- residual + jacobian ----------------
__global__ void k_resjac(const float* camD, const float* xyz, const float* fptr,
                         const float* centre, const float* uv,
                         float* r_out, float* err, float* jc, float* jx,
                         float* initr, unsigned* valid, unsigned* maxbits, int writeJ) {
  int idx = blockIdx.x*blockDim.x + threadIdx.x;
  if (idx >= NPAIRX) return;
  int p = idx / NCX, c = idx % NCX;
  float f = fptr[0];
  float cx = centre[0], cy = centre[1];
  const float* cd = camD + c*64;
  float x = xyz[p*3+0], y = xyz[p*3+1], z = xyz[p*3+2];
  float px = cd[0]*x + cd[3]*y + cd[6]*z + cd[9];
  float py = cd[1]*x + cd[4]*y + cd[7]*z + cd[10];
  float pz = cd[2]*x + cd[5]*y + cd[8]*z + cd[11];
  float iz = 1.f/pz;
  float uvu = uv[idx*2], uvv = uv[idx*2+1];
  float eu = px*iz*f + cx - uvu;
  float ev = py*iz*f + cy - uvv;
  r_out[idx*2]   = eu;
  r_out[idx*2+1] = ev;
  if (initr) { initr[idx*2]=eu; initr[idx*2+1]=ev; }
  bool om = (uvu >= 0.f) && (uvv >= 0.f);
  float er = om ? sqrtf(eu*eu + ev*ev) : 0.f;
  err[idx] = er;
  atomicAdd(valid, om ? 1u : 0u);
  atomicMax(maxbits, __float_as_uint(er));
  if (!writeJ) return;
  float gu  = f*iz;
  float gzu = -f*px*iz*iz;
  float gzv = -f*py*iz*iz;
  float* jrow = jc + (size_t)idx*20;
  for (int k=0;k<6;k++) {
    int j = k>>1;
    float dp0, dp1, dp2;
    if ((k&1)==0) {
      dp0 = x*cd[12+0*3+j] + y*cd[21+0*3+j] + z*cd[39+0*3+j];
      dp1 = x*cd[12+1*3+j] + y*cd[21+1*3+j] + z*cd[39+1*3+j];
      dp2 = x*cd[12+2*3+j] + y*cd[21+2*3+j] + z*cd[39+2*3+j];
    } else {
      dp0 = y*cd[30+0*3+j] + z*cd[48+0*3+j];
      dp1 = y*cd[30+1*3+j] + z*cd[48+1*3+j];
      dp2 = y*cd[30+2*3+j] + z*cd[48+2*3+j];
    }
    jrow[k]    = gu*dp0 + gzu*dp2;
    jrow[10+k] = gu*dp1 + gzv*dp2;
  }
  jrow[6]  = gu;   jrow[16] = 0.f;
  jrow[7]  = 0.f;  jrow[17] = gu;
  jrow[8]  = gzu;  jrow[18] = gzv;
  jrow[9]  = px*iz; jrow[19] = py*iz;
  float* jxr = jx + (size_t)idx*6;
  for (int i=0;i<3;i++) {
    float dp0 = cd[i*3+0], dp1 = cd[i*3+1], dp2 = cd[i*3+2];
    jxr[i]   = gu*dp0 + gzu*dp2;
    jxr[3+i] = gu*dp1 + gzv*dp2;
  }
}

// ---------------- histogram quantile ----------------
__global__ void k_hist(const float* err, const float* scal, unsigned* hist, int stage) {
  int i = blockIdx.x*blockDim.x + threadIdx.x;
  if (i >= NPAIRX) return;
  float v = err[i];
  float range;
  if (stage) { v = fabsf(v - scal[2]); range = scal[9]; }
  else range = __uint_as_float(((const unsigned*)scal)[1]);
  if (!(range > 0.f)) range = 1.f;
  int b = (int)(v / range * (float)NBX);
  if (b < 0) b = 0;
  if (b >= NBX) b = NBX-1;
  atomicAdd(&hist[b], 1u);
}

__global__ void k_qsel(const unsigned* hist, float* scal, int stage) {
  if (threadIdx.x || blockIdx.x) return;
  unsigned valid = ((const unsigned*)scal)[8];
  float q = 1.f - 0.5f * (float)valid / (float)NPAIRX;
  float target = q * (float)(NPAIRX - 1);
  float range = stage ? scal[9] : __uint_as_float(((const unsigned*)scal)[1]);
  if (!(range > 0.f)) range = 1.f;
  float cum = 0.f; int b = 0;
  for (; b < NBX; ++b) { cum += (float)hist[b]; if (cum > target) break; }
  if (b >= NBX) b = NBX-1;
  float val = ((float)b + 0.5f) * range / (float)NBX;
  if (stage == 0) {
    scal[2] = val;
    float mx = __uint_as_float(((const unsigned*)scal)[1]);
    scal[9] = fmaxf(mx - val, val);
  } else {
    scal[3] = val;
    scal[4] = fmaxf(scal[2] + 3.f*val, 5.f);
  }
}

// ---------------- mask + weights + gc + old MSE ----------------
__global__ void k_maskweight(const float* uv, const float* err, const float* rin,
                             const float* jc, const float* scal,
                             float* mask, float* w, float* rw, float* gc, float* mseo) {
  int idx = blockIdx.x*blockDim.x + threadIdx.x;
  if (idx >= NPAIRX) return;
  int c = idx % NCX;
  float cth = scal[4];
  bool om = (uv[idx*2] >= 0.f) && (uv[idx*2+1] >= 0.f);
  float m = (om && err[idx] < cth) ? 1.f : 0.f;
  mask[idx] = m;
  float msum = 0.f;
  const float* jrow = jc + (size_t)idx*20;
  for (int comp=0; comp<2; comp++) {
    float r = rin[idx*2+comp];
    float ra = fabsf(r);
    float hw = (ra <= 3.f) ? 1.f : 3.f*rsqrtf(ra);
    float ww = m*hw;
    w[idx*2+comp] = ww;
    float rr = r*ww;
    rw[idx*2+comp] = rr;
    msum += rr*rr;
    const float* row = jrow + comp*10;
    for (int k=0;k<9;k++) atomicAdd(&gc[c*9+k], row[k]*ww*rr);
    atomicAdd(&gc[72], row[9]*ww*rr);
  }
  atomicAdd(mseo, msum);
}

// ---------------- per-camera JtJ via f32 WMMA (16x16x4 rank-4 updates) ------
// EXEC must be all-1s at every WMMA: loop controls are forced wave-uniform via
// readfirstlane, and the 16->10 column guard is a cndmask (no exec branching).
#define JTJ_SPLIT 8
__global__ void k_jtjcam(const float* jc, const float* w, float* jtjpc) {
  int c = blockIdx.x & (NCX-1);
  int split = blockIdx.x >> 3;
  int wave = __builtin_amdgcn_readfirstlane(threadIdx.x >> 5);
  int lane = threadIdx.x & 31;
  int m = lane & 15, half = lane >> 4;
  v8f acc = {0.f,0.f,0.f,0.f,0.f,0.f,0.f,0.f};
  const int nchunks = (2*NPX)/4;
  int wid = __builtin_amdgcn_readfirstlane(split*8 + wave);
  const int stride = 8*JTJ_SPLIT;
  int mc = (m < 10) ? m : 0;              // clamped column (in-bounds)
  float msel = (m < 10) ? 1.f : 0.f;      // cndmask instead of exec branch
  for (int ch = wid; ch < nchunks; ch += stride) {
    int rbase = ch*4 + half*2;
    v2f a;
#pragma unroll
    for (int kk=0; kk<2; kk++) {
      int row = rbase + kk;
      int pt = row >> 1, comp = row & 1;
      size_t pair = (size_t)pt*NCX + c;
      float val = jc[pair*20 + comp*10 + mc] * msel;
      val *= w[pair*2 + comp];
      a[kk] = val;
    }
    // Jt*J Gram update: A and B hold identical register contents (symmetric).
    acc = __builtin_amdgcn_wmma_f32_16x16x4_f32(false, a, false, a, (short)0, acc, false, false);
  }
  float* dst = jtjpc + c*256;
#pragma unroll
  for (int v=0; v<8; v++)
    atomicAdd(&dst[(v + half*8)*16 + m], acc[v]);
}

// ---------------- scatter per-cam blocks into dense 80x80, damp, copy to S --
__global__ void k_assemble(const float* jtjpc, float* jtj) {
  int tid = blockIdx.x*blockDim.x + threadIdx.x;
  if (tid >= NCX*100) return;
  int c = tid/100, ij = tid%100, i = ij/10, j = ij%10;
  int gi = (i < 9) ? c*9+i : 72;
  int gj = (j < 9) ? c*9+j : 72;
  atomicAdd(&jtj[gi*PADX + gj], jtjpc[c*256 + i*16 + j]);
}

__global__ void k_dampcopy(float* jtj, float* S, const float* scal) {
  int tid = blockIdx.x*blockDim.x + threadIdx.x;
  if (tid >= PADX*PADX) return;
  int r = tid / PADX, cc = tid % PADX;
  float lm = scal[0];
  float v = jtj[tid];
  if (r == cc) v = (r < NCPX) ? fmaxf(v*(1.f+lm), EPX) : 1.f;
  jtj[tid] = v;
  S[tid] = v;
}

// ---------------- per-point: jtj_xyz inverse, gx, E3, BD@gx -----------------
__global__ void k_point(const float* jc, const float* jx, const float* w, const float* rw,
                        const float* scal, float* inv3, float* gx, float* E3, float* bdgx) {
  int p = blockIdx.x*blockDim.x + threadIdx.x;
  if (p >= NPX) return;
  float lm = scal[0];
  float a00=0,a01=0,a02=0,a11=0,a12=0,a22=0;
  float g0=0,g1=0,g2=0;
  float Ef0=0,Ef1=0,Ef2=0;
  float* E3p = E3 + (size_t)p*3*PADX;
  for (int c=0;c<NCX;c++) {
    float Ec[3][9];
    for (int i=0;i<3;i++) for (int k=0;k<9;k++) Ec[i][k]=0.f;
    size_t pair = (size_t)p*NCX + c;
    for (int comp=0; comp<2; comp++) {
      float ww = w[pair*2+comp];
      float rr = rw[pair*2+comp];
      const float* jxr = jx + pair*6 + comp*3;
      float j0 = jxr[0]*ww, j1 = jxr[1]*ww, j2 = jxr[2]*ww;
      a00 += j0*j0; a01 += j0*j1; a02 += j0*j2;
      a11 += j1*j1; a12 += j1*j2; a22 += j2*j2;
      g0 += j0*rr; g1 += j1*rr; g2 += j2*rr;
      const float* row = jc + pair*20 + comp*10;
      for (int k=0;k<9;k++) {
        float jv = row[k]*ww;
        Ec[0][k] += j0*jv; Ec[1][k] += j1*jv; Ec[2][k] += j2*jv;
      }
      float jf = row[9]*ww;
      Ef0 += j0*jf; Ef1 += j1*jf; Ef2 += j2*jf;
    }
    for (int i=0;i<3;i++)
      for (int k=0;k<9;k++) E3p[i*PADX + c*9 + k] = Ec[i][k];
  }
  E3p[0*PADX+72]=Ef0; E3p[1*PADX+72]=Ef1; E3p[2*PADX+72]=Ef2;
  for (int i=0;i<3;i++)
    for (int col=73; col<PADX; col++) E3p[i*PADX+col]=0.f;
  a00 = fmaxf(a00*(1.f+lm), EPX);
  a11 = fmaxf(a11*(1.f+lm), EPX);
  a22 = fmaxf(a22*(1.f+lm), EPX);
  float det = a00*(a11*a22 - a12*a12) - a01*(a01*a22 - a12*a02) + a02*(a01*a12 - a11*a02);
  float id = (fabsf(det) > 1e-30f) ? 1.f/det : 0.f;
  float i00 = (a11*a22 - a12*a12)*id;
  float i01 = (a02*a12 - a01*a22)*id;
  float i02 = (a01*a12 - a02*a11)*id;
  float i11 = (a00*a22 - a02*a02)*id;
  float i12 = (a02*a01 - a00*a12)*id;
  float i22 = (a00*a11 - a01*a01)*id;
  float* iv = inv3 + (size_t)p*9;
  iv[0]=i00; iv[1]=i01; iv[2]=i02; iv[3]=i01; iv[4]=i11; iv[5]=i12; iv[6]=i02; iv[7]=i12; iv[8]=i22;
  gx[p*3+0]=g0; gx[p*3+1]=g1; gx[p*3+2]=g2;
  float h0 = i00*g0 + i01*g1 + i02*g2;
  float h1 = i01*g0 + i11*g1 + i12*g2;
  float h2 = i02*g0 + i12*g1 + i22*g2;
  for (int col=0; col<NCPX; col++) {
    float s = E3p[col]*h0 + E3p[PADX+col]*h1 + E3p[2*PADX+col]*h2;
    atomicAdd(&bdgx[col], s);
  }
}

__global__ void k_rhs(const float* gc, const float* bdgx, float* rhs) {
  int t = threadIdx.x;
  if (t < PADX) rhs[t] = (t < NCPX) ? (gc[t] - bdgx[t]) : 0.f;
}

// ---------------- Schur S -= sum_p E3^T inv3 E3 via f32 WMMA ----------------
__global__ void k_schur(const float* E3, const float* inv3, float* S) {
  int tile = blockIdx.x;
  int ti = tile / 5, tj = tile % 5;
  int wave = __builtin_amdgcn_readfirstlane(threadIdx.x >> 5);
  int lane = threadIdx.x & 31;
  int m = lane & 15, half = lane >> 4;
  int ci = ti*16 + m, cj = tj*16 + m;
  v8f acc = {0.f,0.f,0.f,0.f,0.f,0.f,0.f,0.f};
  for (int p = wave; p < NPX; p += 8) {
    const float* E3p = E3 + (size_t)p*3*PADX;
    const float* iv = inv3 + (size_t)p*9;
    v2f a, b;
    float e0 = E3p[cj], e1 = E3p[PADX+cj], e2 = E3p[2*PADX+cj];
    if (half == 0) {
      a[0] = E3p[ci]; a[1] = E3p[PADX+ci];
      b[0] = iv[0]*e0 + iv[1]*e1 + iv[2]*e2;
      b[1] = iv[3]*e0 + iv[4]*e1 + iv[5]*e2;
    } else {
      a[0] = E3p[2*PADX+ci]; a[1] = 0.f;
      b[0] = iv[6]*e0 + iv[7]*e1 + iv[8]*e2;
      b[1] = 0.f;
    }
    acc = __builtin_amdgcn_wmma_f32_16x16x4_f32(false, a, false, b, (short)0, acc, false, false);
  }
#pragma unroll
  for (int v=0; v<8; v++)
    atomicAdd(&S[(ti*16 + v + half*8)*PADX + tj*16 + m], -acc[v]);
}

// ---------------- dense 80x80 Gauss-Jordan in LDS ---------------------------
__global__ void k_solve(const float* S, const float* rhs, float* dcam) {
  __shared__ float Ls[PADX*PADX];
  __shared__ float lr[PADX];
  __shared__ float pivinv;
  int tid = threadIdx.x;
  for (int i = tid; i < PADX*PADX; i += blockDim.x) Ls[i] = S[i];
  if (tid < PADX) lr[tid] = rhs[tid];
  __syncthreads();
  for (int k=0; k<PADX; k++) {
    if (tid == 0) {
      float pv = Ls[k*PADX+k];
      pivinv = (fabsf(pv) > 1e-20f) ? 1.f/pv : 0.f;
    }
    __syncthreads();
    if (tid < PADX) Ls[k*PADX+tid] *= pivinv;
    if (tid == 0) lr[k] *= pivinv;
    __syncthreads();
    for (int r = tid; r < PADX; r += blockDim.x) {
      if (r == k) continue;
      float fct = Ls[r*PADX+k];
      for (int j=0; j<PADX; j++) Ls[r*PADX+j] -= fct*Ls[k*PADX+j];
      lr[r] -= fct*lr[k];
    }
    __syncthreads();
  }
  if (tid < PADX) {
    float v = lr[tid];
    dcam[tid] = isfinite(v) ? v : 0.f;
  }
}

// ---------------- back-substitute point updates -----------------------------
__global__ void k_dxyz(const float* E3, const float* inv3, const float* gx,
                       const float* dcam, const float* xyz, float* dxyz, float* xyzt) {
  int p = blockIdx.x*blockDim.x + threadIdx.x;
  if (p >= NPX) return;
  const float* E3p = E3 + (size_t)p*3*PADX;
  const float* iv = inv3 + (size_t)p*9;
  float b0 = gx[p*3+0], b1 = gx[p*3+1], b2 = gx[p*3+2];
  for (int col=0; col<NCPX; col++) {
    float dc = dcam[col];
    b0 -= E3p[col]*dc;
    b1 -= E3p[PADX+col]*dc;
    b2 -= E3p[2*PADX+col]*dc;
  }
  for (int j=0; j<3; j++) {
    float dv = b0*iv[j] + b1*iv[3+j] + b2*iv[6+j];
    if (!isfinite(dv)) dv = 0.f;
    dxyz[p*3+j] = dv;
    xyzt[p*3+j] = xyz[p*3+j] - dv;
  }
}

__global__ void k_tmpsmall(const float* cur, const float* dcam, float* tmp) {
  if (threadIdx.x || blockIdx.x) return;
  for (int c=0;c<NCX;c++) {
    for (int k=0;k<6;k++) tmp[c*6+k] = cur[c*6+k] - dcam[c*9+k];
    for (int j=0;j<3;j++) tmp[48+c*3+j] = cur[48+c*3+j] - dcam[c*9+6+j];
  }
  tmp[72] = cur[72] - dcam[72];
}

// ---------------- candidate residual MSE ------------------------------------
__global__ void k_newres(const float* camD, const float* xyz, const float* fptr,
                         const float* centre, const float* uv, const float* mask,
                         float* msen) {
  int idx = blockIdx.x*blockDim.x + threadIdx.x;
  if (idx >= NPAIRX) return;
  int p = idx / NCX, c = idx % NCX;
  float f = fptr[0];
  const float* cd = camD + c*64;
  float x = xyz[p*3], y = xyz[p*3+1], z = xyz[p*3+2];
  float px = cd[0]*x + cd[3]*y + cd[6]*z + cd[9];
  float py = cd[1]*x + cd[4]*y + cd[7]*z + cd[10];
  float pz = cd[2]*x + cd[5]*y + cd[8]*z + cd[11];
  float iz = 1.f/pz;
  float eu = px*iz*f + centre[0] - uv[idx*2];
  float ev = py*iz*f + centre[1] - uv[idx*2+1];
  float m = mask[idx];
  float s = 0.f;
  float ra = fabsf(eu);
  float hw = (ra <= 3.f) ? 1.f : 3.f*rsqrtf(ra);
  float nr = eu*hw*m; s += nr*nr;
  ra = fabsf(ev);
  hw = (ra <= 3.f) ? 1.f : 3.f*rsqrtf(ra);
  nr = ev*hw*m; s += nr*nr;
  atomicAdd(msen, s);
}

__global__ void k_accept(float* cur, const float* tmp, const float* dcam, float* scal) {
  if (threadIdx.x || blockIdx.x) return;
  float sm = ((scal[6] < scal[5]) && (tmp[72] > 0.f)) ? 1.f : 0.f;
  scal[7] = sm;
  scal[0] = scal[0] * (sm/KFX + KFX*(1.f - sm));
  for (int c=0;c<NCX;c++) {
    for (int k=0;k<6;k++) cur[c*6+k] -= sm*dcam[c*9+k];
    for (int j=0;j<3;j++) cur[48+c*3+j] -= sm*dcam[c*9+6+j];
  }
  cur[72] -= sm*dcam[72];
  // re-orthonormalize 6D rotations
  for (int c=0;c<NCX;c++) {
    float b1[3], b2[3];
    gs_b1b2(cur + c*6, b1, b2);
    for (int i=0;i<3;i++) { cur[c*6+i*2] = b1[i]; cur[c*6+i*2+1] = b2[i]; }
  }
}

__global__ void k_applyxyz(float* xyz, const float* dxyz, const float* scal) {
  int p = blockIdx.x*blockDim.x + threadIdx.x;
  if (p >= NPX) return;
  float sm = scal[7];
  for (int j=0;j<3;j++) xyz[p*3+j] -= sm*dxyz[p*3+j];
}

__global__ void k_finalmask(const float* uv, const float* err, const float* scal, float* outmask) {
  int idx = blockIdx.x*blockDim.x + threadIdx.x;
  if (idx >= NPAIRX) return;
  float cth = scal[4];
  bool om = (uv[idx*2] >= 0.f) && (uv[idx*2+1] >= 0.f);
  float m = (om && err[idx] < cth) ? 1.f : 0.f;
  outmask[idx*2] = m;
  outmask[idx*2+1] = m;
}

// ---------------- host orchestration ----------------------------------------
extern "C" void kernel_launch(void* const* d_in, const int* in_sizes, int n_in,
                              void* d_out, int out_size, void* d_ws, size_t ws_size,
                              hipStream_t stream) {
  const float* in_rs = (const float*)d_in[0];
  const float* in_ts = (const float*)d_in[1];
  const float* in_f  = (const float*)d_in[2];
  const float* in_xy = (const float*)d_in[3];
  const float* centre = (const float*)d_in[4];
  const float* uv     = (const float*)d_in[5];
  float* wsf = (float*)d_ws;
  float* out = (float*)d_out;
  float* scal = wsf + O_SCAL;
  unsigned* hist = (unsigned*)(wsf + O_HIST);
  unsigned* validp = ((unsigned*)scal) + 8;
  unsigned* maxbp  = ((unsigned*)scal) + 1;

  hipMemcpyAsync(wsf + O_CUR,      in_rs, 48*sizeof(float), hipMemcpyDeviceToDevice, stream);
  hipMemcpyAsync(wsf + O_CUR + 48, in_ts, 24*sizeof(float), hipMemcpyDeviceToDevice, stream);
  hipMemcpyAsync(wsf + O_CUR + 72, in_f,  sizeof(float),    hipMemcpyDeviceToDevice, stream);
  hipMemcpyAsync(wsf + O_CUR + 73, in_xy, (size_t)NPX*3*sizeof(float), hipMemcpyDeviceToDevice, stream);
  k_init<<<1, 1, 0, stream>>>(scal);

  const int BPAIR = NPAIRX/256, BPT = NPX/256;

  for (int it = 0; it < 3; ++it) {
    hipMemsetAsync(scal + 1, 0, 9*sizeof(float), stream);
    hipMemsetAsync(wsf + O_GC,    0, 80*sizeof(float), stream);
    hipMemsetAsync(wsf + O_BDGX,  0, 80*sizeof(float), stream);
    hipMemsetAsync(wsf + O_JTJPC, 0, 2048*sizeof(float), stream);
    hipMemsetAsync(wsf + O_JTJ,   0, 6400*sizeof(float), stream);

    k_camprep<<<1, 32, 0, stream>>>(wsf + O_CUR, wsf + O_CAMD);
    k_resjac<<<BPAIR, 256, 0, stream>>>(wsf + O_CAMD, wsf + O_CUR + 73, wsf + O_CUR + 72,
                                        centre, uv,
                                        wsf + O_RIN, wsf + O_ERR, wsf + O_JC, wsf + O_JX,
                                        (it == 0) ? (out + OO_RINIT) : nullptr,
                                        validp, maxbp, 1);
    hipMemsetAsync(hist, 0, NBX*sizeof(unsigned), stream);
    k_hist<<<BPAIR, 256, 0, stream>>>(wsf + O_ERR, scal, hist, 0);
    k_qsel<<<1, 1, 0, stream>>>(hist, scal, 0);
    hipMemsetAsync(hist, 0, NBX*sizeof(unsigned), stream);
    k_hist<<<BPAIR, 256, 0, stream>>>(wsf + O_ERR, scal, hist, 1);
    k_qsel<<<1, 1, 0, stream>>>(hist, scal, 1);

    k_maskweight<<<BPAIR, 256, 0, stream>>>(uv, wsf + O_ERR, wsf + O_RIN, wsf + O_JC, scal,
                                            wsf + O_MASK, wsf + O_W, wsf + O_RW,
                                            wsf + O_GC, scal + 5);
    k_jtjcam<<<NCX*JTJ_SPLIT, 256, 0, stream>>>(wsf + O_JC, wsf + O_W, wsf + O_JTJPC);
    k_assemble<<<4, 256, 0, stream>>>(wsf + O_JTJPC, wsf + O_JTJ);
    k_dampcopy<<<25, 256, 0, stream>>>(wsf + O_JTJ, wsf + O_S, scal);
    k_point<<<BPT, 256, 0, stream>>>(wsf + O_JC, wsf + O_JX, wsf + O_W, wsf + O_RW, scal,
                                     wsf + O_INV3, wsf + O_GX, wsf + O_E3, wsf + O_BDGX);
    k_rhs<<<1, 128, 0, stream>>>(wsf + O_GC, wsf + O_BDGX, wsf + O_RHS);
    k_schur<<<25, 256, 0, stream>>>(wsf + O_E3, wsf + O_INV3, wsf + O_S);
    k_solve<<<1, 128, 0, stream>>>(wsf + O_S, wsf + O_RHS, wsf + O_DCAM);
    k_dxyz<<<BPT, 256, 0, stream>>>(wsf + O_E3, wsf + O_INV3, wsf + O_GX, wsf + O_DCAM,
                                    wsf + O_CUR + 73, wsf + O_DXYZ, wsf + O_TMP + 73);
    k_tmpsmall<<<1, 1, 0, stream>>>(wsf + O_CUR, wsf + O_DCAM, wsf + O_TMP);
    k_camprep<<<1, 32, 0, stream>>>(wsf + O_TMP, wsf + O_CAMDT);
    k_newres<<<BPAIR, 256, 0, stream>>>(wsf + O_CAMDT, wsf + O_TMP + 73, wsf + O_TMP + 72,
                                        centre, uv, wsf + O_MASK, scal + 6);
    k_accept<<<1, 1, 0, stream>>>(wsf + O_CUR, wsf + O_TMP, wsf + O_DCAM, scal);
    k_applyxyz<<<BPT, 256, 0, stream>>>(wsf + O_CUR + 73, wsf + O_DXYZ, scal);
  }

  // final residual with last candidate params + final mask
  hipMemsetAsync(scal + 1, 0, 9*sizeof(float), stream);
  k_resjac<<<BPAIR, 256, 0, stream>>>(wsf + O_CAMDT, wsf + O_TMP + 73, wsf + O_TMP + 72,
                                      centre, uv,
                                      out + OO_RFIN, wsf + O_ERR, nullptr, nullptr,
                                      nullptr, validp, maxbp, 0);
  hipMemsetAsync(hist, 0, NBX*sizeof(unsigned), stream);
  k_hist<<<BPAIR, 256, 0, stream>>>(wsf + O_ERR, scal, hist, 0);
  k_qsel<<<1, 1, 0, stream>>>(hist, scal, 0);
  hipMemsetAsync(hist, 0, NBX*sizeof(unsigned), stream);
  k_hist<<<BPAIR, 256, 0, stream>>>(wsf + O_ERR, scal, hist, 1);
  k_qsel<<<1, 1, 0, stream>>>(hist, scal, 1);
  k_finalmask<<<BPAIR, 256, 0, stream>>>(uv, wsf + O_ERR, scal, out + OO_MASK);

  // outputs 0..3: Rs6D, ts, f, xyz (contiguous in ws)
  hipMemcpyAsync(out, wsf + O_CUR, OO_XFER*sizeof(float), hipMemcpyDeviceToDevice, stream);
}